// Conv2D_15891378995436
// MI455X (gfx1250) — compile-verified
//
#include <hip/hip_runtime.h>

typedef __attribute__((ext_vector_type(2))) float v2f;
typedef __attribute__((ext_vector_type(8))) float v8f;
typedef __attribute__((ext_vector_type(4))) unsigned int v4u;
typedef __attribute__((ext_vector_type(4))) int v4i;
typedef __attribute__((ext_vector_type(8))) int v8i;

#define IH 4096
#define IW 4096
#define OH 4090
#define OW 4090
#define KH 7
#define KW 7

#define TILE   64           // 64x64 output region per workgroup
#define IN_T   70           // TILE + KH - 1 rows of input needed
#define XS_LD  76           // LDS row stride in words == TDM tile_dim0
                            // (even; 76 mod 64 = 12 -> conflict-free ds_load_b64)
#define NCHUNK 6            // K chunks of 4 per ky  (K = 24 >= 22)

__global__ __launch_bounds__(256)
void conv7x7_wmma_f32(const float* __restrict__ X,
                      const float* __restrict__ Wt,
                      const float* __restrict__ bias,
                      float* __restrict__ out) {
  __shared__ float Xs[IN_T * XS_LD];          // 70*76*4  = 21280 B (TDM dest)
  __shared__ v2f   Btab[KH * NCHUNK * 32];    // 42*32*8  = 10752 B

  const int tid  = threadIdx.x;
  const int lane = tid & 31;
  const int wave = tid >> 5;
  const int Y0   = blockIdx.y * TILE;
  const int X0   = blockIdx.x * TILE;

  // ---- Tensor Data Mover: DMA the 70x76 input tile global -> LDS ----
  // D# per CDNA5 ISA 8.3-8.6. tensor_dim = remaining extent from the tile's
  // global_addr so hardware zero-fills out-of-image reads (edge tiles).
  // tile_dim0 = 76 words == LDS row stride -> contiguous TDM writes
  // reproduce the padded Xs layout exactly.
  if (tid < 32) {                 // one wave issues the TDM op
    const int remx = IW - X0;     // 64..4096 (fits 16 bits)
    const int remy = IH - Y0;

    v4u g0;
    g0[0] = 1u;                                   // count=1, user mode
    g0[1] = (unsigned int)(unsigned long long)(uintptr_t)Xs;  // lds_addr
    unsigned long long ga =
        (unsigned long long)(uintptr_t)X +
        ((unsigned long long)Y0 * IW + (unsigned long long)X0) * 4ull;
    g0[2] = (unsigned int)ga;                     // global_addr[31:0]
    g0[3] = (unsigned int)((ga >> 32) & 0x01FFFFFFull) | 0x80000000u; // type=2

    v8i g1;
    g1[0] = (2 << 16);                            // data_size=4B; no mask/pad
    g1[1] = (remx & 0xFFFF) << 16;                // tensor_dim0[15:0]
    g1[2] = ((remx >> 16) & 0xFFFF) | ((remy & 0xFFFF) << 16); // dim0.hi|dim1.lo
    g1[3] = ((remy >> 16) & 0xFFFF) | (XS_LD << 16);           // dim1.hi|tile_dim0
    g1[4] = IN_T;                                 // tile_dim1=70, tile_dim2=0
    g1[5] = IW;                                   // tensor_dim0_stride[31:0]
    g1[6] = 0;                                    // stride0.hi | stride1.lo
    g1[7] = 0;                                    // stride1.hi

    v4i g2 = {};                                  // 2-D tile: groups 2/3 unused
    v4i g3 = {};

#if defined(__clang_major__) && (__clang_major__ >= 23)
    v8i g4 = {};
    __builtin_amdgcn_tensor_load_to_lds(g0, g1, g2, g3, g4, 0);
#else
    __builtin_amdgcn_tensor_load_to_lds(g0, g1, g2, g3, 0);
#endif
  }

  // ---- build banded-Toeplitz B fragments (overlaps with TDM) ----
  // B_ky[j][n] = w[ky][j-n] for 0<=j-n<7 else 0 ; per-lane WMMA layout:
  // lanes 0-15 hold K = j0+0 (v0), j0+1 (v1); lanes 16-31 hold K = j0+2, j0+3
  for (int e = tid; e < KH * NCHUNK * 32; e += 256) {
    int ky  = e / (NCHUNK * 32);
    int rem = e - ky * (NCHUNK * 32);
    int c   = rem >> 5;
    int l   = rem & 31;
    int hh  = l >> 4;
    int n   = l & 15;
    int j   = c * 4 + 2 * hh;
    int d0  = j - n;
    int d1  = j + 1 - n;
    v2f b;
    b.x = (d0 >= 0 && d0 < KW) ? Wt[ky * KW + d0] : 0.0f;
    b.y = (d1 >= 0 && d1 < KW) ? Wt[ky * KW + d1] : 0.0f;
    Btab[e] = b;
  }

  if (tid < 32) __builtin_amdgcn_s_wait_tensorcnt(0);  // tile landed in LDS
  __syncthreads();

  const float bv = bias[0];
  const int h = lane >> 4;      // lane half: selects K pair (A/B) and M+8 (C/D)
  const int m = lane & 15;      // A: row M ; C/D: column N

  // 8 waves x 2 tiles = 16 tiles of 16x16 covering the 64x64 region
  for (int t = wave * 2; t < wave * 2 + 2; ++t) {
    const int ty = t >> 2;
    const int tx = t & 3;
    v8f acc = {};

    for (int ky = 0; ky < KH; ++ky) {
      const float* arow =
          &Xs[(ty * 16 + m + ky) * XS_LD + tx * 16 + 2 * h];
      const v2f* brow = &Btab[ky * NCHUNK * 32 + lane];
#pragma unroll
      for (int c = 0; c < NCHUNK; ++c) {
        v2f a = *(const v2f*)(arow + c * 4);   // ds_load_b64 (8B aligned)
        v2f b = brow[c * 32];                  // ds_load_b64
        acc = __builtin_amdgcn_wmma_f32_16x16x4_f32(
            /*neg_a=*/false, a, /*neg_b=*/false, b,
            /*c_mod=*/(short)0, acc, /*reuse_a=*/false, /*reuse_b=*/false);
      }
    }

    // ---- store: C/D layout lane n = column, VGPR r -> row r + 8*h ----
    const int col = X0 + tx * 16 + m;
    if (col < OW) {
      const int rowbase = Y0 + ty * 16 + 8 * h;
#pragma unroll
      for (int r = 0; r < 8; ++r) {
        int row = rowbase + r;
        if (row < OH) out[row * OW + col] = acc[r] + bv;
      }
    }
  }
}

extern "C" void kernel_launch(void* const* d_in, const int* in_sizes, int n_in,
                              void* d_out, int out_size, void* d_ws, size_t ws_size,
                              hipStream_t stream) {
  const float* X    = (const float*)d_in[0];
  const float* Wt   = (const float*)d_in[1];
  const float* bias = (const float*)d_in[2];
  float* out        = (float*)d_out;

  dim3 grid((OW + TILE - 1) / TILE, (OH + TILE - 1) / TILE);  // 64 x 64
  conv7x7_wmma_f32<<<grid, dim3(256), 0, stream>>>(X, Wt, bias, out);
}